// filter_trans_37383395344761
// MI455X (gfx1250) — compile-verified
//
#include <hip/hip_runtime.h>

// db2 dec_lo reversed (matches reference _DB2_DEC_LO_REV), fp32 like reference
#define K0 0.48296291314469025f
#define K1 0.836516303737469f
#define K2 0.22414386804185735f
#define K3 (-0.12940952255092145f)

typedef __attribute__((ext_vector_type(2))) float v2f;
typedef __attribute__((ext_vector_type(8))) float v8f;

// pywt 'symmetric' half-sample extension (pad=3) composed with the xe[1:] shift.
// y[n] = sum_j K_j * x[emap(2n+1+j)]
__device__ __forceinline__ constexpr int emap(int m, int L) {
  return (m < 3) ? (2 - m) : ((m <= L + 2) ? (m - 3) : (2 * L + 2 - m));
}

// ---------------------------------------------------------------------------
// Pass along a NON-contiguous axis: view tensor as (OUTER, L, INNER), filter
// the middle dim. All dims compile-time -> divmod becomes magic-multiply,
// all indexing 32-bit. Consecutive threads walk INNER -> fully coalesced.
// 4 loads + 1 store + 4 FMA per output: memory-roofline bound.
// ---------------------------------------------------------------------------
template <int OUTER, int L, int O, int INNER>
__global__ __launch_bounds__(256) void dwt_low_strided_t(
    const float* __restrict__ x, float* __restrict__ y) {
  constexpr unsigned total = (unsigned)OUTER * O * INNER;
  const unsigned tid = blockIdx.x * 256u + threadIdx.x;
  if (tid >= total) return;
  const unsigned i = tid % (unsigned)INNER;        // const divisor -> mulhi
  const unsigned t = tid / (unsigned)INNER;
  const unsigned n = t % (unsigned)O;
  const unsigned o = t / (unsigned)O;
  const float* xb = x + o * (unsigned)(L * INNER) + i;
  const int m = 2 * (int)n + 1;
  float acc =       K0 * xb[emap(m,     L) * INNER];
  acc = fmaf(K1, xb[emap(m + 1, L) * INNER], acc);
  acc = fmaf(K2, xb[emap(m + 2, L) * INNER], acc);
  acc = fmaf(K3, xb[emap(m + 3, L) * INNER], acc);
  y[tid] = acc;
}

// ---------------------------------------------------------------------------
// Pass along the LAST (contiguous) axis as a banded matmul via
// V_WMMA_F32_16X16X4_F32:  Y(16x16) = sum_ch A(16x4) x B(4x16).
//
// Each wave owns one output-column tile (ct) and iterates TPW row tiles.
// The 9 B fragments (sparse filter-matrix columns incl. symmetric boundary
// folding) depend only on ct: synthesized once into 18 VGPRs and reused.
// Inner loop: 9 b64 A-loads straight from global (consecutive chunks reuse
// the same L0 lines) overlapped with 9 WMMAs; interior tiles take a uniform
// fast path of 8 unconditional stores off one base address with immediate
// offsets (O is a template constant). No LDS, no barriers; every branch
// around the WMMAs is wave-uniform so EXEC is all-1s.
// ---------------------------------------------------------------------------
template <int L, int O, int CT, int TPW>
__global__ __launch_bounds__(256) void dwt_low_lastaxis_wmma_t(
    const float* __restrict__ x, float* __restrict__ y,
    int R, int rtg_count) {
  const int lane = threadIdx.x & 31;
  const int wave = threadIdx.x >> 5;
  const int id = blockIdx.x * 8 + wave;
  if (id >= CT * rtg_count) return;          // wave-uniform; no barriers used

  const int ct  = id % CT;
  const int rtg = id / CT;
  const int cb  = ct * 16;
  const int kstart = max(0, 2 * cb - 2);     // input window [kstart, kstart+35]
  const int half = lane >> 4;                // 0: K rows {0,1}; 1: K rows {2,3}
  const int nl   = lane & 15;                // A row / output column within tile
  const int nn   = cb + nl;

  // --- B fragments for all 9 k-chunks (once per wave) ---
  int e0 = -1000, e1 = -1000, e2 = -1000, e3 = -1000;
  if (nn < O) {
    const int m = 2 * nn + 1;
    e0 = emap(m, L); e1 = emap(m + 1, L); e2 = emap(m + 2, L); e3 = emap(m + 3, L);
  }
  v2f bf[9];
#pragma unroll
  for (int ch = 0; ch < 9; ++ch) {
    const int m0 = kstart + 4 * ch + 2 * half;
    const int m1 = m0 + 1;
    bf[ch].x = ((e0 == m0) ? K0 : 0.f) + ((e1 == m0) ? K1 : 0.f) +
               ((e2 == m0) ? K2 : 0.f) + ((e3 == m0) ? K3 : 0.f);
    bf[ch].y = ((e0 == m1) ? K0 : 0.f) + ((e1 == m1) ? K1 : 0.f) +
               ((e2 == m1) ? K2 : 0.f) + ((e3 == m1) ? K3 : 0.f);
  }

  // --- iterate row tiles, reusing B ---
  for (int t = 0; t < TPW; ++t) {
    const int rb = (rtg * TPW + t) * 16;
    if (rb >= R) break;                      // wave-uniform

    const int arow  = min(rb + nl, R - 1);   // clamped rows feed only unstored D rows
    const int abase = arow * L;              // 32-bit; const-mul by L

    v8f c = {};
#pragma unroll
    for (int ch = 0; ch < 9; ++ch) {
      // clamp so col and col+1 stay in-range; clamped extras hit B == 0 only if L is even
      const int col = min(kstart + 4 * ch + 2 * half, L - 2);
      v2f a;
      a.x = x[abase + col];
      a.y = x[abase + col + 1];
      c = __builtin_amdgcn_wmma_f32_16x16x4_f32(
          /*neg_a=*/false, a, /*neg_b=*/false, bf[ch],
          /*c_mod=*/(short)0, c, /*reuse_a=*/false, /*reuse_b=*/false);
    }

    // D layout: VGPR v -> row rb + v + 8*half, col nn. Coalesced along columns.
    if (rb + 16 <= R && cb + 16 <= O) {
      // interior tile: one base address, immediate offsets v*O*4
      const int base = (rb + 8 * half) * O + nn;
#pragma unroll
      for (int v = 0; v < 8; ++v)
        y[base + v * O] = c[v];
    } else {
#pragma unroll
      for (int v = 0; v < 8; ++v) {
        const int row = rb + v + 8 * half;
        if (row < R && nn < O)
          y[row * O + nn] = c[v];
      }
    }
  }
}

// ---------------------------------------------------------------------------
// 6-pass pipeline, shapes fixed by the reference:
//   in (2,4,128,224,224) -A-> (..65,224,224) -B-> (..65,113,224)
//   -C-> (..65,113,113)  -D-> (..34,113,113) -E-> (..34,58,113) -F-> (..34,58,58)
// ws0 = 26,091,520 floats, ws1 = 13,162,240 floats (157 MB total; the big
// intermediates stay resident in the 192 MB L2; memory floor ~13 us at
// 23.3 TB/s, compute is negligible even with the banded-as-dense WMMA).
// ---------------------------------------------------------------------------
extern "C" void kernel_launch(void* const* d_in, const int* in_sizes, int n_in,
                              void* d_out, int out_size, void* d_ws, size_t ws_size,
                              hipStream_t stream) {
  (void)in_sizes; (void)n_in; (void)out_size; (void)ws_size;
  const float* x = (const float*)d_in[0];
  float* out = (float*)d_out;
  float* ws0 = (float*)d_ws;
  float* ws1 = ws0 + 26091520LL;

  // Pass A: axis 2, 128 -> 65; outer=8, inner=224*224
  {
    constexpr long long total = 8LL * 65 * 50176;
    dwt_low_strided_t<8, 128, 65, 50176>
        <<<(int)((total + 255) / 256), 256, 0, stream>>>(x, ws0);
  }
  // Pass B: axis 3, 224 -> 113 on (8,65,224,224); outer=8*65, inner=224
  {
    constexpr long long total = 520LL * 113 * 224;
    dwt_low_strided_t<520, 224, 113, 224>
        <<<(int)((total + 255) / 256), 256, 0, stream>>>(ws0, ws1);
  }
  // Pass C: last axis, 224 -> 113 on (8,65,113,224); R=58760 (WMMA)
  {
    constexpr int R = 58760, TPW = 8, CT = 8;        // ceil(113/16)=8
    constexpr int rtg = (((R + 15) / 16) + TPW - 1) / TPW;  // 460
    constexpr int waves = CT * rtg;                  // 3680
    dwt_low_lastaxis_wmma_t<224, 113, CT, TPW>
        <<<(waves + 7) / 8, 256, 0, stream>>>(ws1, ws0, R, rtg);
  }
  // Pass D: axis 2, 65 -> 34 on (8,65,113,113); outer=8, inner=113*113
  {
    constexpr long long total = 8LL * 34 * 12769;
    dwt_low_strided_t<8, 65, 34, 12769>
        <<<(int)((total + 255) / 256), 256, 0, stream>>>(ws0, ws1);
  }
  // Pass E: axis 3, 113 -> 58 on (8,34,113,113); outer=8*34, inner=113
  {
    constexpr long long total = 272LL * 58 * 113;
    dwt_low_strided_t<272, 113, 58, 113>
        <<<(int)((total + 255) / 256), 256, 0, stream>>>(ws1, ws0);
  }
  // Pass F: last axis, 113 -> 58 on (8,34,58,113); R=15776 (WMMA) -> d_out
  {
    constexpr int R = 15776, TPW = 4, CT = 4;        // ceil(58/16)=4
    constexpr int rtg = (((R + 15) / 16) + TPW - 1) / TPW;  // 247
    constexpr int waves = CT * rtg;                  // 988
    dwt_low_lastaxis_wmma_t<113, 58, CT, TPW>
        <<<(waves + 7) / 8, 256, 0, stream>>>(ws0, out, R, rtg);
  }
}